// Model_10917806867251
// MI455X (gfx1250) — compile-verified
//
#include <hip/hip_runtime.h>
#include <hip/hip_bf16.h>
#include <math.h>

typedef __attribute__((ext_vector_type(2))) float v2f;
typedef __attribute__((ext_vector_type(8))) float v8f;

#define N_NODES    100000
#define N_EDGES    800000
#define NUM_GRAPHS 512
#define FEAT 64
#define EMB  128
#define GAMMA 0.5f

// ---------------- fill ----------------
__global__ void k_fill0(float* p, long n) {
  long i = (long)blockIdx.x * blockDim.x + threadIdx.x;
  if (i < n) p[i] = 0.f;
}

// ---------------- WMMA GEMM: C = act((A [+A2]) @ W + bias) ----------------
// One wave per 16x(16*NT) C macro-tile. A-slice (K<=128) cached in VGPRs once,
// reused across NT n-tiles -> A traffic cut NT x vs per-tile scheme.
// v_wmma_f32_16x16x4_f32 per-lane layout (ISA 7.12.2):
//   A (16x4): lane L -> row L%16, (a.x,a.y) = cols (k+2h, k+2h+1), h = L/16
//   B (4x16): lane L -> col L%16, (b.x,b.y) = rows (k+2h, k+2h+1)
//   D (16x16): acc[r], lane<16 -> row m0+r, lane>=16 -> row m0+8+r
template<bool RELU, int K, int NT>
__global__ void k_gemm16(const float* __restrict__ A, const float* __restrict__ A2,
                         const float* __restrict__ W, const float* __restrict__ bias,
                         float* __restrict__ C, int N) {
  const int lane = threadIdx.x;
  const int half = lane >> 4, l16 = lane & 15;
  const int mA = blockIdx.x * 16 + l16;
  const int n0 = blockIdx.y * (16 * NT);
  const float* arow = A + (long)mA * K;

  float areg[K / 2];
  #pragma unroll
  for (int kk = 0; kk < K / 4; ++kk) {
    const int ka = kk * 4 + half * 2;
    areg[2 * kk]     = arow[ka];
    areg[2 * kk + 1] = arow[ka + 1];
  }
  if (A2) {
    const float* arow2 = A2 + (long)mA * K;
    #pragma unroll
    for (int kk = 0; kk < K / 4; ++kk) {
      const int ka = kk * 4 + half * 2;
      areg[2 * kk]     += arow2[ka];
      areg[2 * kk + 1] += arow2[ka + 1];
    }
  }

  v8f acc[NT];
  #pragma unroll
  for (int t = 0; t < NT; ++t) acc[t] = (v8f){0.f,0.f,0.f,0.f,0.f,0.f,0.f,0.f};

  #pragma unroll
  for (int kk = 0; kk < K / 4; ++kk) {
    const int ka = kk * 4 + half * 2;
    v2f a;
    a.x = areg[2 * kk];
    a.y = areg[2 * kk + 1];
    const float* w0 = W + (long)ka * N + n0 + l16;
    const float* w1 = W + (long)(ka + 1) * N + n0 + l16;
    #pragma unroll
    for (int t = 0; t < NT; ++t) {
      v2f b;
      b.x = w0[t * 16];
      b.y = w1[t * 16];
      acc[t] = __builtin_amdgcn_wmma_f32_16x16x4_f32(false, a, false, b, (short)0,
                                                     acc[t], false, false);
    }
  }

  const int row0 = blockIdx.x * 16 + half * 8;
  #pragma unroll
  for (int t = 0; t < NT; ++t) {
    const int nB = n0 + t * 16 + l16;
    const float bv = bias ? bias[nB] : 0.f;
    #pragma unroll
    for (int r = 0; r < 8; ++r) {
      float v = acc[t][r] + bv;
      if (RELU) v = fmaxf(v, 0.f);
      C[(long)(row0 + r) * N + nB] = v;
    }
  }
}

// ---------------- edge scatter: msg[dst] += h[src] ----------------
__global__ void k_edge_scatter(const float* __restrict__ h, const int* __restrict__ src,
                               const int* __restrict__ dst, float* __restrict__ msg,
                               int nE) {
  long tid = (long)blockIdx.x * blockDim.x + threadIdx.x;
  int e = (int)(tid >> 5), lane = (int)(tid & 31);
  if (e >= nE) return;
  const int s = src[e], d = dst[e];
  const float* hs = h + (long)s * EMB;
  float* md = msg + (long)d * EMB;
  #pragma unroll
  for (int c = lane; c < EMB; c += 32) atomicAdd(&md[c], hs[c]);
}

// ---------------- gate pass 1: column sum / sumsq of G = Hrat @ W1a ----------------
// (b1 and the broadcast pos/neg concat columns are per-column constants -> they
//  cancel exactly in BatchNorm, so only Hrat @ W1[:128,:] matters.)
// A-slice cached in VGPRs; one wave sweeps all 16 n-tiles of the 256 columns.
__global__ void k_gate_stats(const float* __restrict__ Hrat, const float* __restrict__ W1,
                             float* __restrict__ csum, float* __restrict__ csq) {
  const int lane = threadIdx.x;
  const int half = lane >> 4, l16 = lane & 15;
  const int mA = blockIdx.x * 16 + l16;
  const float* arow = Hrat + (long)mA * EMB;

  float areg[EMB / 2];
  #pragma unroll
  for (int kk = 0; kk < EMB / 4; ++kk) {
    const int ka = kk * 4 + half * 2;
    areg[2 * kk]     = arow[ka];
    areg[2 * kk + 1] = arow[ka + 1];
  }

  for (int nt = 0; nt < 16; ++nt) {
    const int nB = nt * 16 + l16;
    v8f acc = {0.f,0.f,0.f,0.f,0.f,0.f,0.f,0.f};
    #pragma unroll
    for (int kk = 0; kk < EMB / 4; ++kk) {
      const int ka = kk * 4 + half * 2;
      v2f a, b;
      a.x = areg[2 * kk];
      a.y = areg[2 * kk + 1];
      b.x = W1[(long)ka * 256 + nB];
      b.y = W1[(long)(ka + 1) * 256 + nB];
      acc = __builtin_amdgcn_wmma_f32_16x16x4_f32(false, a, false, b, (short)0, acc,
                                                  false, false);
    }
    float s1 = 0.f, s2 = 0.f;
    #pragma unroll
    for (int r = 0; r < 8; ++r) { float z = acc[r]; s1 += z; s2 += z * z; }
    s1 += __shfl_xor(s1, 16, 32);  // combine the two 8-row halves of column nB
    s2 += __shfl_xor(s2, 16, 32);
    if (lane < 16) { atomicAdd(&csum[nB], s1); atomicAdd(&csq[nB], s2); }
  }
}

// ---------------- gate pass 2: gate = sigmoid(relu(bn(G)) @ W2 + b2) ----------------
__global__ void k_gate_finish(const float* __restrict__ Hrat, const float* __restrict__ W1,
                              const float* __restrict__ csum, const float* __restrict__ csq,
                              const float* __restrict__ bng, const float* __restrict__ bnb,
                              const float* __restrict__ W2, const float* __restrict__ b2,
                              float* __restrict__ gate, int M) {
  const int lane = threadIdx.x;
  const int half = lane >> 4, l16 = lane & 15;
  const int mA = blockIdx.x * 16 + l16;
  const float* arow = Hrat + (long)mA * EMB;
  const float invM = 1.f / (float)M;

  float areg[EMB / 2];
  #pragma unroll
  for (int kk = 0; kk < EMB / 4; ++kk) {
    const int ka = kk * 4 + half * 2;
    areg[2 * kk]     = arow[ka];
    areg[2 * kk + 1] = arow[ka + 1];
  }

  float part[8];
  #pragma unroll
  for (int r = 0; r < 8; ++r) part[r] = 0.f;

  for (int nt = 0; nt < 16; ++nt) {
    const int nB = nt * 16 + l16;
    v8f acc = {0.f,0.f,0.f,0.f,0.f,0.f,0.f,0.f};
    #pragma unroll
    for (int kk = 0; kk < EMB / 4; ++kk) {
      const int ka = kk * 4 + half * 2;
      v2f a, b;
      a.x = areg[2 * kk];
      a.y = areg[2 * kk + 1];
      b.x = W1[(long)ka * 256 + nB];
      b.y = W1[(long)(ka + 1) * 256 + nB];
      acc = __builtin_amdgcn_wmma_f32_16x16x4_f32(false, a, false, b, (short)0, acc,
                                                  false, false);
    }
    const float mean = csum[nB] * invM;
    const float var  = csq[nB] * invM - mean * mean;
    const float sc   = bng[nB] * rsqrtf(var + 1e-5f);
    const float sh   = bnb[nB];
    const float w2v  = W2[nB];
    #pragma unroll
    for (int r = 0; r < 8; ++r) {
      float v = fmaxf((acc[r] - mean) * sc + sh, 0.f);
      part[r] += v * w2v;
    }
  }
  // reduce over the 16 lanes within each half (each half shares 8 rows)
  #pragma unroll
  for (int r = 0; r < 8; ++r) {
    part[r] += __shfl_xor(part[r], 8, 16);
    part[r] += __shfl_xor(part[r], 4, 16);
    part[r] += __shfl_xor(part[r], 2, 16);
    part[r] += __shfl_xor(part[r], 1, 16);
  }
  if (l16 == 0) {
    const int row0 = blockIdx.x * 16 + half * 8;
    const float bb = b2[0];
    #pragma unroll
    for (int r = 0; r < 8; ++r)
      gate[row0 + r] = 1.f / (1.f + expf(-(part[r] + bb)));
  }
}

// ---------------- node scatter: h_r/h_env/r_num/e_num ----------------
__global__ void k_node_scatter(const float* __restrict__ h, const float* __restrict__ gate,
                               const int* __restrict__ gind, float* __restrict__ hr,
                               float* __restrict__ henv, float* __restrict__ rnum,
                               float* __restrict__ enm, int nNodes) {
  long tid = (long)blockIdx.x * blockDim.x + threadIdx.x;
  int node = (int)(tid >> 5), lane = (int)(tid & 31);
  if (node >= nNodes) return;
  const int g = gind[node];
  const float gt = gate[node];
  const float* hn = h + (long)node * EMB;
  #pragma unroll
  for (int c = lane; c < EMB; c += 32) {
    float v = hn[c];
    atomicAdd(&hr[(long)g * EMB + c], gt * v);
    atomicAdd(&henv[(long)g * EMB + c], (1.f - gt) * v);
  }
  if (lane == 0) { atomicAdd(&rnum[g], gt); atomicAdd(&enm[g], 1.f - gt); }
}

// ---------------- per-column mean/var (biased) of [M,N] ----------------
__global__ void k_colstats(const float* __restrict__ X, float* __restrict__ mean,
                           float* __restrict__ var, int M, int N) {
  int n = blockIdx.x * blockDim.x + threadIdx.x;
  if (n >= N) return;
  float s = 0.f, q = 0.f;
  for (int m = 0; m < M; ++m) { float v = X[(long)m * N + n]; s += v; q += v * v; }
  float mu = s / (float)M;
  mean[n] = mu;
  var[n] = q / (float)M - mu * mu;
}

// ---------------- P/Q prep for pred_rep: bn = P[i]+Q[j] pre-relu ----------------
// h_rep[i,j]@W1 = Ar0[i] + Ae0[j] + const;  BN over the full i x j cross product
// has mean = mAr+mAe+const, var = vAr+vAe  =>  bn = (Ar0-mAr)*sc + (Ae0-mAe)*sc + beta
__global__ void k_prep_pq(const float* __restrict__ Ar, const float* __restrict__ Ae,
                          const float* __restrict__ mAr, const float* __restrict__ vAr,
                          const float* __restrict__ mAe, const float* __restrict__ vAe,
                          const float* __restrict__ bng, const float* __restrict__ bnb,
                          float* __restrict__ P, float* __restrict__ Q) {
  long i = (long)blockIdx.x * blockDim.x + threadIdx.x;
  if (i >= (long)NUM_GRAPHS * 256) return;
  int n = (int)(i & 255);
  float sc = bng[n] * rsqrtf(vAr[n] + vAe[n] + 1e-5f);
  P[i] = (Ar[i] - mAr[n]) * sc;
  Q[i] = (Ae[i] - mAe[n]) * sc + bnb[n];
}

// ---------------- pred_rep: out[i*512+j] = relu(P[i]+Q[j]) @ W2 + b2 ----------------
__global__ void k_pred_rep(const float* __restrict__ P, const float* __restrict__ Q,
                           const float* __restrict__ W2, const float* __restrict__ b2,
                           float* __restrict__ out) {
  const int i = blockIdx.x;
  const int w = threadIdx.x >> 5, lane = threadIdx.x & 31;
  const int j = blockIdx.y * 8 + w;
  const float* Pi = P + (long)i * 256;
  const float* Qj = Q + (long)j * 256;
  float a0 = 0.f, a1 = 0.f;
  #pragma unroll
  for (int n = lane; n < 256; n += 32) {
    float v = fmaxf(Pi[n] + Qj[n], 0.f);
    a0 += v * W2[2 * n];
    a1 += v * W2[2 * n + 1];
  }
  #pragma unroll
  for (int m = 16; m > 0; m >>= 1) {
    a0 += __shfl_xor(a0, m, 32);
    a1 += __shfl_xor(a1, m, 32);
  }
  if (lane == 0) {
    long o = ((long)i * NUM_GRAPHS + j) * 2;
    out[o] = a0 + b2[0];
    out[o + 1] = a1 + b2[1];
  }
}

// ---------------- pred_rem (512 rows; BN stats = stats of Ar alone) ----------------
__global__ void k_pred_rem(const float* __restrict__ Ar, const float* __restrict__ mAr,
                           const float* __restrict__ vAr, const float* __restrict__ bng,
                           const float* __restrict__ bnb, const float* __restrict__ W2,
                           const float* __restrict__ b2, float* __restrict__ out) {
  const int i = blockIdx.x;
  const int lane = threadIdx.x;
  float a0 = 0.f, a1 = 0.f;
  #pragma unroll
  for (int n = lane; n < 256; n += 32) {
    float sc = bng[n] * rsqrtf(vAr[n] + 1e-5f);
    float v = fmaxf((Ar[(long)i * 256 + n] - mAr[n]) * sc + bnb[n], 0.f);
    a0 += v * W2[2 * n];
    a1 += v * W2[2 * n + 1];
  }
  #pragma unroll
  for (int m = 16; m > 0; m >>= 1) {
    a0 += __shfl_xor(a0, m, 32);
    a1 += __shfl_xor(a1, m, 32);
  }
  if (lane == 0) { out[2 * i] = a0 + b2[0]; out[2 * i + 1] = a1 + b2[1]; }
}

// ---------------- loss_reg ----------------
__global__ void k_loss(const float* __restrict__ rnum, const float* __restrict__ enm,
                       float* __restrict__ out) {
  __shared__ float sh[NUM_GRAPHS];
  int t = threadIdx.x;
  float r = rnum[t] + 1e-8f, e = enm[t] + 1e-8f;
  sh[t] = fabsf(r / (r + e) - GAMMA);
  __syncthreads();
  for (int s = NUM_GRAPHS / 2; s > 0; s >>= 1) {
    if (t < s) sh[t] += sh[t + s];
    __syncthreads();
  }
  if (t == 0) out[0] = sh[0] / (float)NUM_GRAPHS;
}

// =====================================================================
extern "C" void kernel_launch(void* const* d_in, const int* in_sizes, int n_in,
                              void* d_out, int out_size, void* d_ws, size_t ws_size,
                              hipStream_t stream) {
  const float* nodes    = (const float*)d_in[0];
  const int*   e_src    = (const int*)d_in[2];
  const int*   e_dst    = (const int*)d_in[3];
  const int*   gind     = (const int*)d_in[4];
  const float* W_enc_in = (const float*)d_in[8];
  const float* W_enc    = (const float*)d_in[9];
  const float* b_enc    = (const float*)d_in[10];
  const float* W_rat_in = (const float*)d_in[11];
  const float* W_rat    = (const float*)d_in[12];
  const float* b_rat    = (const float*)d_in[13];
  const float* gate_W1  = (const float*)d_in[14];
  const float* gate_bng = (const float*)d_in[16];
  const float* gate_bnb = (const float*)d_in[17];
  const float* gate_W2  = (const float*)d_in[18];
  const float* gate_b2  = (const float*)d_in[19];
  const float* pred_W1  = (const float*)d_in[20];
  const float* pred_bng = (const float*)d_in[22];
  const float* pred_bnb = (const float*)d_in[23];
  const float* pred_W2  = (const float*)d_in[24];
  const float* pred_b2  = (const float*)d_in[25];
  float* out = (float*)d_out;

  // ---- workspace carve-up (floats) ----
  float* ws   = (float*)d_ws;
  float* q0   = ws;                         // [100000,128]
  float* q1   = q0 + (long)N_NODES * EMB;   // [100000,128]
  float* msg  = q1 + (long)N_NODES * EMB;   // [100000,128]
  float* csum = msg + (long)N_NODES * EMB;  // [256]
  float* csq  = csum + 256;                 // [256]
  float* gate = csq + 256;                  // [100000]
  float* hr   = gate + N_NODES;             // [512,128]
  float* henv = hr + NUM_GRAPHS * EMB;      // [512,128]
  float* rnum = henv + NUM_GRAPHS * EMB;    // [512]
  float* enm  = rnum + NUM_GRAPHS;          // [512]
  float* Ar   = enm + NUM_GRAPHS;           // [512,256]
  float* Ae   = Ar + NUM_GRAPHS * 256;      // [512,256]
  float* mAr  = Ae + NUM_GRAPHS * 256;      // [256] x4
  float* vAr  = mAr + 256;
  float* mAe  = vAr + 256;
  float* vAe  = mAe + 256;
  float* P    = vAe + 256;                  // [512,256]
  float* Q    = P + NUM_GRAPHS * 256;       // [512,256]

  const dim3 b32(32), b256(256);
  const long hN = (long)N_NODES * EMB;
  const unsigned fillB = (unsigned)((hN + 255) / 256);
  const unsigned edgeB = (unsigned)(((long)N_EDGES * 32 + 255) / 256);
  const unsigned nodeB = (unsigned)(((long)N_NODES * 32 + 255) / 256);

  // ---- 1) rationale encoder (2 layers) -> h_rat in q0 ----
  k_gemm16<true, FEAT, 4><<<dim3(N_NODES / 16, EMB / 64), b32, 0, stream>>>(
      nodes, nullptr, W_rat_in, nullptr, q0, EMB);
  for (int i = 0; i < 2; ++i) {
    const float* hin = (i == 0) ? q0 : q1;
    float* hout      = (i == 0) ? q1 : q0;
    k_fill0<<<fillB, b256, 0, stream>>>(msg, hN);
    k_edge_scatter<<<edgeB, b256, 0, stream>>>(hin, e_src, e_dst, msg, N_EDGES);
    k_gemm16<true, EMB, 4><<<dim3(N_NODES / 16, EMB / 64), b32, 0, stream>>>(
        hin, msg, W_rat + (long)i * EMB * EMB, b_rat + i * EMB, hout, EMB);
  }

  // ---- 2) gate MLP (BN cancels b1 and the broadcast pos/neg columns) ----
  k_fill0<<<2, b256, 0, stream>>>(csum, 512);  // csum+csq contiguous
  k_gate_stats<<<dim3(N_NODES / 16), b32, 0, stream>>>(q0, gate_W1, csum, csq);
  k_gate_finish<<<dim3(N_NODES / 16), b32, 0, stream>>>(
      q0, gate_W1, csum, csq, gate_bng, gate_bnb, gate_W2, gate_b2, gate, N_NODES);

  // ---- 3) query encoder (3 layers) -> h_node in q1 ----
  k_gemm16<true, FEAT, 4><<<dim3(N_NODES / 16, EMB / 64), b32, 0, stream>>>(
      nodes, nullptr, W_enc_in, nullptr, q0, EMB);
  for (int i = 0; i < 3; ++i) {
    const float* hin = (i % 2 == 0) ? q0 : q1;
    float* hout      = (i % 2 == 0) ? q1 : q0;
    k_fill0<<<fillB, b256, 0, stream>>>(msg, hN);
    k_edge_scatter<<<edgeB, b256, 0, stream>>>(hin, e_src, e_dst, msg, N_EDGES);
    k_gemm16<true, EMB, 4><<<dim3(N_NODES / 16, EMB / 64), b32, 0, stream>>>(
        hin, msg, W_enc + (long)i * EMB * EMB, b_enc + i * EMB, hout, EMB);
  }

  // ---- 4) gated pooling ----
  long poolN = 2L * NUM_GRAPHS * EMB + 2L * NUM_GRAPHS;  // hr,henv,rnum,enm contiguous
  k_fill0<<<(unsigned)((poolN + 255) / 256), b256, 0, stream>>>(hr, poolN);
  k_node_scatter<<<nodeB, b256, 0, stream>>>(q1, gate, gind, hr, henv, rnum, enm, N_NODES);

  // ---- 5) pred head: Ar = h_r @ W1[:128], Ae = h_env @ W1[:128] ----
  k_gemm16<false, EMB, 4><<<dim3(NUM_GRAPHS / 16, 256 / 64), b32, 0, stream>>>(
      hr, nullptr, pred_W1, nullptr, Ar, 256);
  k_gemm16<false, EMB, 4><<<dim3(NUM_GRAPHS / 16, 256 / 64), b32, 0, stream>>>(
      henv, nullptr, pred_W1, nullptr, Ae, 256);
  k_colstats<<<1, b256, 0, stream>>>(Ar, mAr, vAr, NUM_GRAPHS, 256);
  k_colstats<<<1, b256, 0, stream>>>(Ae, mAe, vAe, NUM_GRAPHS, 256);
  k_prep_pq<<<NUM_GRAPHS, b256, 0, stream>>>(Ar, Ae, mAr, vAr, mAe, vAe,
                                             pred_bng, pred_bnb, P, Q);

  // ---- 6) outputs: pred_rep [512*512,2], pred_rem [512,2], loss ----
  k_pred_rep<<<dim3(NUM_GRAPHS, NUM_GRAPHS / 8), b256, 0, stream>>>(
      P, Q, pred_W2, pred_b2, out);
  k_pred_rem<<<NUM_GRAPHS, b32, 0, stream>>>(
      Ar, mAr, vAr, pred_bng, pred_bnb, pred_W2, pred_b2,
      out + (long)NUM_GRAPHS * NUM_GRAPHS * 2);
  k_loss<<<1, NUM_GRAPHS, 0, stream>>>(
      rnum, enm, out + (long)NUM_GRAPHS * NUM_GRAPHS * 2 + NUM_GRAPHS * 2);
}